// e3nnEncoder_55405078118471
// MI455X (gfx1250) — compile-verified
//
// MI455X (gfx1250, CDNA5) implementation of the e3nn CG encoder.
//
// Design rationale (compile-only; reasoned from MI455X specs):
//  * Total model FLOPs ~30 GF -> far below matrix-op ceilings; the binding
//    resource is HBM traffic. The per-edge tensor-product weights (up to
//    384 f32/edge over 200K edges) would cost ~1.8 GB of HBM round-trips if
//    materialized; instead each conv kernel FUSES the 36->36->numel fc MLP
//    (done with V_WMMA_F32_16X16X4_F32, wave32, M=16 edges per wave) with the
//    w3j tensor product and the atomic segment-scatter, keeping weights in
//    LDS (320KB/WGP on CDNA5 makes the ~36KB/block staging trivial).
//  * f32 WMMA (16x16x4) keeps reference precision; K dims (36/84/256) are
//    multiples of 4. Dense head (2500x84 @ 84x256, tanh, @ 256x256) is a
//    plain WMMA GEMM.
//  * Segment means: unsafeAtomicAdd (global_atomic_add_f32) + degree divide.
//  * All state lives in d_ws and is re-initialized every call; all launches
//    on `stream` (graph-capture safe, deterministic work).
//
// Workspace use: ~43 MB.

#include <hip/hip_runtime.h>
#include <hip/hip_bf16.h>
#include <math.h>

typedef __attribute__((ext_vector_type(2))) float v2f;
typedef __attribute__((ext_vector_type(8))) float v8f;

#define NSC        12
#define N_ATOMS_C  20000
#define N_CG_C     2500
#define N_NBR_C    100000
#define N_CGNBR_C  15000
#define NODE_STR   48      // padded node feature stride
#define CDIV(a,b)  (((a)+(b)-1)/(b))

// ---------------------------------------------------------------- WMMA ----
__device__ __forceinline__ v8f wmma4(v2f a, v2f b, v8f c) {
  // D(16x16,f32) += A(16x4,f32) x B(4x16,f32)  -- CDNA5 V_WMMA_F32_16X16X4_F32
  return __builtin_amdgcn_wmma_f32_16x16x4_f32(false, a, false, b, (short)0, c,
                                               false, false);
}

// ------------------------------------------------------------- helpers ----
__device__ __forceinline__ void sph9(float rx, float ry, float rz, float* sh) {
  float n   = sqrtf(rx*rx + ry*ry + rz*rz);
  float inv = 1.0f / fmaxf(n, 1e-6f);
  const float S3 = 1.7320508075688772f;
  // reference ordering u[[1,2,0]] = (y, z, x)
  float y0 = S3 * ry * inv;   // y
  float y1 = S3 * rz * inv;   // z
  float y2 = S3 * rx * inv;   // x
  sh[0] = 1.0f; sh[1] = y0; sh[2] = y1; sh[3] = y2;
  const float A  = 0.31622776601683794f;   // 1/sqrt(10)
  const float Cc = 0.18257418583505536f;   // A/sqrt(3)
  const float Y2 = 2.0412414523193150f;    // sqrt(5)/|3*C112[z,z,:]|
  sh[4] = Y2 * 2.0f * A * y2 * y0;                     // xy
  sh[5] = Y2 * 2.0f * A * y0 * y1;                     // yz
  sh[6] = Y2 * Cc * (2.0f*y1*y1 - y0*y0 - y2*y2);      // 2z^2-x^2-y^2
  sh[7] = Y2 * 2.0f * A * y2 * y1;                     // xz
  sh[8] = Y2 * A * (y2*y2 - y0*y0);                    // x^2-y^2
}

template<int N>
__device__ __forceinline__ void gaussN(float d, float stop, float* out) {
  float step  = stop / (float)(N - 1);
  float coeff = -0.5f / (step * step);
#pragma unroll
  for (int j = 0; j < N; ++j) { float t = d - step * (float)j; out[j] = __expf(coeff * t * t); }
}

template<int FI>
__device__ __forceinline__ void mlp12(const float* __restrict__ w1, const float* __restrict__ b1,
                                      const float* __restrict__ w2, const float* __restrict__ b2,
                                      const float* in, float* out12) {
  float h[12];
#pragma unroll
  for (int j = 0; j < 12; ++j) {
    float s = b1[j];
#pragma unroll
    for (int i = 0; i < FI; ++i) s += in[i] * w1[i*12 + j];
    h[j] = fmaxf(s, 0.0f);
  }
#pragma unroll
  for (int j = 0; j < 12; ++j) {
    float s = b2[j];
#pragma unroll
    for (int i = 0; i < 12; ++i) s += h[i] * w2[i*12 + j];
    out12[j] = s;
  }
}

// ----------------------------------------------------- tensor-product -----
// path kinds: 0:(0,0,0) 1:(0,1,1) 2:(1,0,1) 3:(1,1,0) 4:(1,1,1) 5:(1,2,1)
struct TpPath { int o1, o3, m1, m3, kind, wo; float f; };

#define ISQ12 0.2886751345948129f
#define ISQ16 0.25f
#define ISQ20 0.22360679774997896f
#define ISQ4  0.5f
#define ISQ24 0.20412414523193154f

__constant__ TpPath c_tp[18] = {
  // layer 0 (off 0, n=2)
  {0,0,12,12,0,0,ISQ12}, {0,12,12,4,1,144,ISQ12},
  // layer 1 (off 2, n=6)
  {0,0,12,12,0,0,ISQ16}, {0,12,12,4,1,144,ISQ20}, {12,12,4,4,2,192,ISQ20},
  {12,0,4,12,3,208,ISQ16}, {12,24,4,4,4,256,ISQ4}, {12,12,4,4,5,272,ISQ20},
  // layer 2 (off 8, n=10)
  {0,0,12,12,0,0,ISQ16}, {0,12,12,4,1,144,ISQ24}, {12,12,4,4,2,192,ISQ24},
  {12,0,4,12,3,208,ISQ16}, {12,24,4,4,4,256,ISQ12}, {12,12,4,4,5,272,ISQ24},
  {24,24,4,4,2,288,ISQ12}, {24,12,4,4,4,304,ISQ24}, {24,36,4,12,3,320,ISQ4},
  {24,24,4,4,5,368,ISQ12}
};

// ------------------------------------------------------- small kernels ----
__global__ void fill0_kernel(float* p, long n) {
  long t = (long)blockIdx.x * blockDim.x + threadIdx.x;
  if (t < n) p[t] = 0.0f;
}

__global__ void edge_intra_kernel(const int* __restrict__ nbr, int nraw,
                                  const float* __restrict__ xyz, const int* __restrict__ zz,
                                  const float* __restrict__ w1, const float* __restrict__ b1,
                                  const float* __restrict__ w2, const float* __restrict__ b2,
                                  float maxr, int* __restrict__ src, int* __restrict__ dst,
                                  float* __restrict__ deg, float* __restrict__ sh_out,
                                  float* __restrict__ e_out) {
  int e = blockIdx.x * blockDim.x + threadIdx.x;
  int E = 2 * nraw;
  if (e >= E) return;
  int s, d;
  if (e < nraw) { s = nbr[2*e];         d = nbr[2*e + 1]; }
  else          { int q = e - nraw; s = nbr[2*q + 1]; d = nbr[2*q]; }
  src[e] = s; dst[e] = d;
  unsafeAtomicAdd(&deg[s], 1.0f);
  float rx = xyz[3*d]   - xyz[3*s];
  float ry = xyz[3*d+1] - xyz[3*s+1];
  float rz = xyz[3*d+2] - xyz[3*s+2];
  float sh[9]; sph9(rx, ry, rz, sh);
#pragma unroll
  for (int k = 0; k < 9; ++k) sh_out[(size_t)e*9 + k] = sh[k];
  float in[14];
  in[0] = (float)zz[s]; in[1] = (float)zz[d];
  in[2] = 0.f; in[3] = 0.f; in[4] = 0.f; in[5] = 0.f;
  gaussN<8>(sqrtf(rx*rx + ry*ry + rz*rz), maxr, in + 6);
  float o[12]; mlp12<14>(w1, b1, w2, b2, in, o);
#pragma unroll
  for (int k = 0; k < 12; ++k) e_out[(size_t)e*12 + k] = o[k];
}

__global__ void edge_cross_kernel(const int* __restrict__ mapping,
                                  const float* __restrict__ xyz, const float* __restrict__ cg_xyz,
                                  const float* __restrict__ w1, const float* __restrict__ b1,
                                  const float* __restrict__ w2, const float* __restrict__ b2,
                                  float maxr, float* __restrict__ sh_out,
                                  float* __restrict__ e_out) {
  int e = blockIdx.x * blockDim.x + threadIdx.x;
  if (e >= N_ATOMS_C) return;
  int j = mapping[e];
  float rx = xyz[3*e]   - cg_xyz[3*j];
  float ry = xyz[3*e+1] - cg_xyz[3*j+1];
  float rz = xyz[3*e+2] - cg_xyz[3*j+2];
  float sh[9]; sph9(rx, ry, rz, sh);
#pragma unroll
  for (int k = 0; k < 9; ++k) sh_out[(size_t)e*9 + k] = sh[k];
  float in[8];
  gaussN<8>(sqrtf(rx*rx + ry*ry + rz*rz), maxr, in);
  float o[12]; mlp12<8>(w1, b1, w2, b2, in, o);
#pragma unroll
  for (int k = 0; k < 12; ++k) e_out[(size_t)e*12 + k] = o[k];
}

__global__ void map_count_kernel(const int* __restrict__ mapping, float* __restrict__ cnt) {
  int i = blockIdx.x * blockDim.x + threadIdx.x;
  if (i < N_ATOMS_C) unsafeAtomicAdd(&cnt[mapping[i]], 1.0f);
}

__global__ void init_node_kernel(const float* __restrict__ emb, const int* __restrict__ zz,
                                 int n, float* __restrict__ node) {
  int t = blockIdx.x * blockDim.x + threadIdx.x;
  if (t >= n * NODE_STR) return;
  int i = t / NODE_STR, k = t % NODE_STR;
  node[t] = (k < NSC) ? emb[zz[i]*NSC + k] : 0.0f;
}

__global__ void update_node_kernel(float* __restrict__ next, const float* __restrict__ cur,
                                   const float* __restrict__ accIntra, const float* __restrict__ degIntra,
                                   const float* __restrict__ accInter, const float* __restrict__ degInter,
                                   int n, int din, int dout) {
  int t = blockIdx.x * blockDim.x + threadIdx.x;
  if (t >= n * dout) return;
  int i = t / dout, k = t % dout;
  float v = (k < din) ? cur[(size_t)i*NODE_STR + k] : 0.0f;
  v += accIntra[(size_t)i*NODE_STR + k] / fmaxf(degIntra[i], 1.0f);
  float d2 = degInter ? fmaxf(degInter[i], 1.0f) : 1.0f;
  v += accInter[(size_t)i*NODE_STR + k] / d2;
  next[(size_t)i*NODE_STR + k] = v;
}

__global__ void pool_scatter_kernel(const float* __restrict__ anode, const int* __restrict__ mapping,
                                    float* __restrict__ pooled) {
  int t = blockIdx.x * blockDim.x + threadIdx.x;
  if (t >= N_ATOMS_C * 48) return;
  int i = t / 48, k = t % 48;
  unsafeAtomicAdd(&pooled[(size_t)mapping[i]*84 + k], anode[(size_t)i*NODE_STR + k]);
}

__global__ void pool_final_kernel(float* __restrict__ pooled, const float* __restrict__ cnode,
                                  const float* __restrict__ cnt) {
  int t = blockIdx.x * blockDim.x + threadIdx.x;
  if (t >= N_CG_C * 84) return;
  int j = t / 84, k = t % 84;
  if (k < 48) pooled[t] = pooled[t] / fmaxf(cnt[j], 1.0f);
  else        pooled[t] = cnode[(size_t)j*NODE_STR + (k - 48)];
}

// ----------------------------------------- fused fc-MLP + TP + scatter ----
// One wave (32 lanes) per block; 16 edges per block.
//   fcin (16x36) -> WMMA -> relu hidden (16x36) -> WMMA -> w (16 x numel) in
//   LDS -> per-lane tensor product -> atomic scatter.
__global__ void __launch_bounds__(32)
conv_tp_kernel(int E, int layer, int din, int numel, int dout,
               const float* __restrict__ ef, const float* __restrict__ esh,
               const float* __restrict__ nA, const int* __restrict__ idxA,
               const float* __restrict__ nB, const int* __restrict__ idxB,
               const float* __restrict__ X1, const int* __restrict__ idxX1,
               const float* __restrict__ w1, const float* __restrict__ b1,
               const float* __restrict__ w2, const float* __restrict__ b2,
               float* __restrict__ acc, const int* __restrict__ scat) {
  __shared__ float sFc [16][40];
  __shared__ float sHid[16][48];
  __shared__ float sW  [16][384];
  __shared__ float sX1 [16][40];
  __shared__ float sOut[16][48];

  const int lane = threadIdx.x;
  const int e0   = blockIdx.x * 16;
  const int m    = lane & 15;
  const int half = lane >> 4;

  // stage fc input: [edgefeat(12) | nodeA[:12] | nodeB[:12]]
  for (int t = lane; t < 16*36; t += 32) {
    int r = t / 36, c = t % 36, e = e0 + r;
    float v = 0.0f;
    if (e < E) {
      if (c < 12)      v = ef[(size_t)e*12 + c];
      else if (c < 24) { int ia = idxA ? idxA[e] : e; v = nA[(size_t)ia*NODE_STR + (c - 12)]; }
      else             { int ib = idxB ? idxB[e] : e; v = nB[(size_t)ib*NODE_STR + (c - 24)]; }
    }
    sFc[r][c] = v;
  }
  // stage x1 = node[dst]
  for (int t = lane; t < 16*din; t += 32) {
    int r = t / din, c = t % din, e = e0 + r;
    sX1[r][c] = (e < E) ? X1[(size_t)(idxX1 ? idxX1[e] : e)*NODE_STR + c] : 0.0f;
  }
  for (int t = lane; t < 16*48; t += 32) sOut[t/48][t%48] = 0.0f;
  if (lane == 0 && e0 + 16 < E) __builtin_prefetch(ef + (size_t)(e0 + 16)*12, 0, 0);
  __syncthreads();

  // GEMM1: hidden = relu(fcin @ w1 + b1)   (M=16 edges, K=36, N=36->48)
  for (int nt = 0; nt < 3; ++nt) {
    v8f a8 = {0.f,0.f,0.f,0.f,0.f,0.f,0.f,0.f};
    const int col = nt*16 + m;
#pragma unroll
    for (int k = 0; k < 9; ++k) {
      int kb = k*4 + 2*half;
      v2f a; a.x = sFc[m][kb]; a.y = sFc[m][kb+1];
      v2f b;
      b.x = (col < 36) ? w1[(size_t)(kb+0)*36 + col] : 0.0f;
      b.y = (col < 36) ? w1[(size_t)(kb+1)*36 + col] : 0.0f;
      a8 = wmma4(a, b, a8);
    }
#pragma unroll
    for (int i = 0; i < 8; ++i) {
      int row = i + 8*half;
      float v = (col < 36) ? fmaxf(a8[i] + b1[col], 0.0f) : 0.0f;
      sHid[row][col] = v;
    }
  }
  __syncthreads();

  // GEMM2: w = hidden @ w2 + b2   (N = numel, multiple of 16)
  const int ntiles = numel >> 4;
  for (int nt = 0; nt < ntiles; ++nt) {
    v8f a8 = {0.f,0.f,0.f,0.f,0.f,0.f,0.f,0.f};
    const int col = nt*16 + m;
#pragma unroll
    for (int k = 0; k < 9; ++k) {
      int kb = k*4 + 2*half;
      v2f a; a.x = sHid[m][kb]; a.y = sHid[m][kb+1];
      v2f b; b.x = w2[(size_t)(kb+0)*numel + col]; b.y = w2[(size_t)(kb+1)*numel + col];
      a8 = wmma4(a, b, a8);
    }
#pragma unroll
    for (int i = 0; i < 8; ++i) sW[i + 8*half][col] = a8[i] + b2[col];
  }
  __syncthreads();

  // tensor product: lane e handles edge e0+lane
  if (lane < 16) {
    int e = e0 + lane;
    if (e < E) {
      const float* shp = esh + (size_t)e*9;
      float s0=shp[0], s1=shp[1], s2=shp[2], s3=shp[3], s4=shp[4],
            s5=shp[5], s6=shp[6], s7=shp[7], s8=shp[8];
      const float* X = sX1[lane];
      const float* W = sW[lane];
      float*       O = sOut[lane];
      int off = (layer == 0) ? 0 : (layer == 1 ? 2 : 8);
      int np  = (layer == 0) ? 2 : (layer == 1 ? 6 : 10);
      const float IS3 = 0.5773502691896258f, IS6 = 0.4082482904638631f;
      const float A = 0.31622776601683794f, Cc = 0.18257418583505536f;
      for (int p = 0; p < np; ++p) {
        TpPath pp = c_tp[off + p];
        int o1 = pp.o1, o3 = pp.o3, m1 = pp.m1, m3 = pp.m3, wo = pp.wo;
        float f = pp.f;
        switch (pp.kind) {
          case 0:
            for (int u = 0; u < m1; ++u) {
              float xu = X[o1+u] * s0 * f;
              for (int wv = 0; wv < m3; ++wv) O[o3+wv] += W[wo + u*m3 + wv] * xu;
            } break;
          case 1:
            for (int u = 0; u < m1; ++u) {
              float g = X[o1+u] * f * IS3;
              for (int wv = 0; wv < m3; ++wv) {
                float ww = W[wo + u*m3 + wv] * g;
                O[o3+3*wv+0] += ww*s1; O[o3+3*wv+1] += ww*s2; O[o3+3*wv+2] += ww*s3;
              }
            } break;
          case 2:
            for (int u = 0; u < m1; ++u) {
              float g  = f * IS3 * s0;
              float x0 = X[o1+3*u]*g, x1 = X[o1+3*u+1]*g, x2 = X[o1+3*u+2]*g;
              for (int wv = 0; wv < m3; ++wv) {
                float ww = W[wo + u*m3 + wv];
                O[o3+3*wv+0] += ww*x0; O[o3+3*wv+1] += ww*x1; O[o3+3*wv+2] += ww*x2;
              }
            } break;
          case 3:
            for (int u = 0; u < m1; ++u) {
              float du = (X[o1+3*u]*s1 + X[o1+3*u+1]*s2 + X[o1+3*u+2]*s3) * IS3 * f;
              for (int wv = 0; wv < m3; ++wv) O[o3+wv] += W[wo + u*m3 + wv] * du;
            } break;
          case 4:
            for (int u = 0; u < m1; ++u) {
              float x0 = X[o1+3*u], x1 = X[o1+3*u+1], x2 = X[o1+3*u+2];
              float c0 = (x1*s3 - x2*s2) * IS6 * f;
              float c1 = (x2*s1 - x0*s3) * IS6 * f;
              float c2 = (x0*s2 - x1*s1) * IS6 * f;
              for (int wv = 0; wv < m3; ++wv) {
                float ww = W[wo + u*m3 + wv];
                O[o3+3*wv+0] += ww*c0; O[o3+3*wv+1] += ww*c1; O[o3+3*wv+2] += ww*c2;
              }
            } break;
          default: // kind 5: (1,2,1) via C112
            for (int u = 0; u < m1; ++u) {
              float x0 = X[o1+3*u], x1 = X[o1+3*u+1], x2 = X[o1+3*u+2];
              float c0 = (A*(x2*s4 + x1*s5) - x0*(Cc*s6 + A*s8)) * f;
              float c1 = (A*(x0*s5 + x2*s7) + 2.0f*Cc*x1*s6)     * f;
              float c2 = (A*(x0*s4 + x1*s7) + x2*(A*s8 - Cc*s6)) * f;
              for (int wv = 0; wv < m3; ++wv) {
                float ww = W[wo + u*m3 + wv];
                O[o3+3*wv+0] += ww*c0; O[o3+3*wv+1] += ww*c1; O[o3+3*wv+2] += ww*c2;
              }
            } break;
        }
      }
    }
  }
  __syncthreads();

  // scatter-add into segment accumulator
  for (int t = lane; t < 16*dout; t += 32) {
    int r = t / dout, c = t % dout, e = e0 + r;
    if (e < E) {
      int tgt = scat ? scat[e] : e;
      unsafeAtomicAdd(&acc[(size_t)tgt*NODE_STR + c], sOut[r][c]);
    }
  }
}

// ----------------------------------------------------- dense head GEMM ----
// out[M,N] = act(A[M,K] @ W[K,N] + bias) ; one wave per 16x16 tile.
__global__ void __launch_bounds__(32)
dense_wmma_kernel(const float* __restrict__ Am, int M, int K,
                  const float* __restrict__ W, const float* __restrict__ bias,
                  float* __restrict__ out, int N, int do_tanh) {
  const int lane = threadIdx.x;
  const int m    = lane & 15;
  const int half = lane >> 4;
  const int row0 = blockIdx.x * 16;
  const int col  = blockIdx.y * 16 + m;
  const int row  = row0 + m;
  v8f a8 = {0.f,0.f,0.f,0.f,0.f,0.f,0.f,0.f};
  const int ks = K >> 2;
  for (int k = 0; k < ks; ++k) {
    int kb = k*4 + 2*half;
    v2f a;
    a.x = (row < M) ? Am[(size_t)row*K + kb]     : 0.0f;
    a.y = (row < M) ? Am[(size_t)row*K + kb + 1] : 0.0f;
    v2f b;
    b.x = W[(size_t)(kb+0)*N + col];
    b.y = W[(size_t)(kb+1)*N + col];
    a8 = wmma4(a, b, a8);
  }
#pragma unroll
  for (int i = 0; i < 8; ++i) {
    int r = row0 + i + 8*half;
    if (r < M) {
      float v = a8[i] + bias[col];
      out[(size_t)r*N + col] = do_tanh ? tanhf(v) : v;
    }
  }
}

// ------------------------------------------------------------- launch -----
extern "C" void kernel_launch(void* const* d_in, const int* in_sizes, int n_in,
                              void* d_out, int out_size, void* d_ws, size_t ws_size,
                              hipStream_t stream) {
  (void)in_sizes; (void)n_in; (void)out_size; (void)ws_size;
  const int*   z       = (const int*)  d_in[0];
  const float* xyz     = (const float*)d_in[1];
  const int*   cg_z    = (const int*)  d_in[2];
  const float* cg_xyz  = (const float*)d_in[3];
  const int*   mapping = (const int*)  d_in[4];
  const int*   nbr     = (const int*)  d_in[5];
  const int*   cg_nbr  = (const int*)  d_in[6];
  // params flattened alphabetically (jax tree order), each MLP as b1,b2,w1,w2
  const float* prm[66];
  for (int i = 0; i < 66; ++i) prm[i] = (const float*)d_in[7 + i];
  // 0..3 atom_edge{b1,b2,w1,w2}  4 atom_emb  5..8 cg_edge  9 cg_emb
  // 10..13 cross_edge  14..17 dense{b1,b2,w1,w2}
  // 18/22/26 fc_a2c_l  30/34/38 fc_atom_l  42/46/50 fc_c2a_l  54/58/62 fc_cg_l

  // ---- workspace layout ----
  char* wp = (char*)d_ws;
  auto alloc = [&](size_t nbytes) { void* p = wp; wp += (nbytes + 255) & ~(size_t)255; return p; };
  const int EA = 2 * N_NBR_C, EC = 2 * N_CGNBR_C;
  int*   asrc   = (int*)  alloc((size_t)EA * 4);
  int*   adst   = (int*)  alloc((size_t)EA * 4);
  int*   csrc   = (int*)  alloc((size_t)EC * 4);
  int*   cdst   = (int*)  alloc((size_t)EC * 4);
  float* a_deg  = (float*)alloc((size_t)N_ATOMS_C * 4);
  float* c_deg  = (float*)alloc((size_t)N_CG_C * 4);
  float* mapcnt = (float*)alloc((size_t)N_CG_C * 4);
  float* a_sh   = (float*)alloc((size_t)EA * 9 * 4);
  float* a_edge = (float*)alloc((size_t)EA * 12 * 4);
  float* c_sh   = (float*)alloc((size_t)EC * 9 * 4);
  float* c_edge = (float*)alloc((size_t)EC * 12 * 4);
  float* x_sh   = (float*)alloc((size_t)N_ATOMS_C * 9 * 4);
  float* x_edge = (float*)alloc((size_t)N_ATOMS_C * 12 * 4);
  float* aN[2]; aN[0] = (float*)alloc((size_t)N_ATOMS_C * NODE_STR * 4);
                aN[1] = (float*)alloc((size_t)N_ATOMS_C * NODE_STR * 4);
  float* cN[2]; cN[0] = (float*)alloc((size_t)N_CG_C * NODE_STR * 4);
                cN[1] = (float*)alloc((size_t)N_CG_C * NODE_STR * 4);
  float* acc_a  = (float*)alloc((size_t)N_ATOMS_C * NODE_STR * 4);
  float* acc_ai = (float*)alloc((size_t)N_ATOMS_C * NODE_STR * 4);
  float* acc_c  = (float*)alloc((size_t)N_CG_C * NODE_STR * 4);
  float* acc_ci = (float*)alloc((size_t)N_CG_C * NODE_STR * 4);
  float* pooled = (float*)alloc((size_t)N_CG_C * 84 * 4);
  float* hbuf   = (float*)alloc((size_t)N_CG_C * 256 * 4);

  // ---- degrees / counts ----
  {
    long n = N_ATOMS_C + 2 * N_CG_C;  // a_deg, c_deg, mapcnt adjacent; 256B padding leaves mapcnt tail unzeroed
    fill0_kernel<<<CDIV(n,256),256,0,stream>>>(a_deg, n);
  }
  // ---- edge features (+ src/dst build, + degree counting) ----
  edge_intra_kernel<<<CDIV(EA,256),256,0,stream>>>(nbr, N_NBR_C, xyz, z,
      prm[2], prm[0], prm[3], prm[1], 12.0f, asrc, adst, a_deg, a_sh, a_edge);
  edge_intra_kernel<<<CDIV(EC,256),256,0,stream>>>(cg_nbr, N_CGNBR_C, cg_xyz, cg_z,
      prm[7], prm[5], prm[8], prm[6], 30.0f, csrc, cdst, c_deg, c_sh, c_edge);
  edge_cross_kernel<<<CDIV(N_ATOMS_C,256),256,0,stream>>>(mapping, xyz, cg_xyz,
      prm[12], prm[10], prm[13], prm[11], 30.0f, x_sh, x_edge);
  map_count_kernel<<<CDIV(N_ATOMS_C,256),256,0,stream>>>(mapping, mapcnt);

  // ---- node embeddings ----
  init_node_kernel<<<CDIV(N_ATOMS_C*NODE_STR,256),256,0,stream>>>(prm[4], z, N_ATOMS_C, aN[0]);
  init_node_kernel<<<CDIV(N_CG_C*NODE_STR,256),256,0,stream>>>(prm[9], cg_z, N_CG_C, cN[0]);

  const int din_[3]   = {12, 24, 36};
  const int dout_[3]  = {24, 36, 48};
  const int numel_[3] = {192, 288, 384};
  int ai = 0, ci = 0;
  for (int l = 0; l < 3; ++l) {
    // zero all conv accumulators (contiguous region)
    long accn = (long)(N_ATOMS_C*2 + N_CG_C*2) * NODE_STR;
    fill0_kernel<<<CDIV(accn,256),256,0,stream>>>(acc_a, accn);

    const float* fA = prm[30 + 4*l + 2]; const float* fAb1 = prm[30 + 4*l + 0];
    const float* fAw2 = prm[30 + 4*l + 3]; const float* fAb2 = prm[30 + 4*l + 1];
    // atom intra: x1 = a_node[adst], scatter by asrc
    conv_tp_kernel<<<CDIV(EA,16),32,0,stream>>>(EA, l, din_[l], numel_[l], dout_[l],
        a_edge, a_sh, aN[ai], asrc, aN[ai], adst, aN[ai], adst,
        fA, fAb1, fAw2, fAb2, acc_a, asrc);
    // c2a: x1 = c_node[mapping], scatter identity (count 1)
    conv_tp_kernel<<<CDIV(N_ATOMS_C,16),32,0,stream>>>(N_ATOMS_C, l, din_[l], numel_[l], dout_[l],
        x_edge, x_sh, aN[ai], nullptr, cN[ci], mapping, cN[ci], mapping,
        prm[42+4*l+2], prm[42+4*l+0], prm[42+4*l+3], prm[42+4*l+1], acc_ai, nullptr);
    if (l < 2) {
      // cg intra
      conv_tp_kernel<<<CDIV(EC,16),32,0,stream>>>(EC, l, din_[l], numel_[l], dout_[l],
          c_edge, c_sh, cN[ci], csrc, cN[ci], cdst, cN[ci], cdst,
          prm[54+4*l+2], prm[54+4*l+0], prm[54+4*l+3], prm[54+4*l+1], acc_c, csrc);
      // a2c: x1 = a_node[e], scatter by mapping
      conv_tp_kernel<<<CDIV(N_ATOMS_C,16),32,0,stream>>>(N_ATOMS_C, l, din_[l], numel_[l], dout_[l],
          x_edge, x_sh, aN[ai], nullptr, cN[ci], mapping, aN[ai], nullptr,
          prm[18+4*l+2], prm[18+4*l+0], prm[18+4*l+3], prm[18+4*l+1], acc_ci, mapping);
    }
    update_node_kernel<<<CDIV(N_ATOMS_C*dout_[l],256),256,0,stream>>>(
        aN[1-ai], aN[ai], acc_a, a_deg, acc_ai, nullptr, N_ATOMS_C, din_[l], dout_[l]);
    if (l < 2) {
      update_node_kernel<<<CDIV(N_CG_C*dout_[l],256),256,0,stream>>>(
          cN[1-ci], cN[ci], acc_c, c_deg, acc_ci, mapcnt, N_CG_C, din_[l], dout_[l]);
      ci ^= 1;
    }
    ai ^= 1;
  }

  // ---- pooling + dense head ----
  fill0_kernel<<<CDIV(N_CG_C*84,256),256,0,stream>>>(pooled, (long)N_CG_C*84);
  pool_scatter_kernel<<<CDIV(N_ATOMS_C*48,256),256,0,stream>>>(aN[ai], mapping, pooled);
  pool_final_kernel<<<CDIV(N_CG_C*84,256),256,0,stream>>>(pooled, cN[ci], mapcnt);

  dim3 g1(CDIV(N_CG_C,16), 256/16);
  dense_wmma_kernel<<<g1,32,0,stream>>>(pooled, N_CG_C, 84,  prm[16], prm[14], hbuf, 256, 1);
  dense_wmma_kernel<<<g1,32,0,stream>>>(hbuf,   N_CG_C, 256, prm[17], prm[15], (float*)d_out, 256, 0);
}